// EvalEig_64914135712399
// MI455X (gfx1250) — compile-verified
//
#include <hip/hip_runtime.h>
#include <stdint.h>

// Problem constants (match reference)
#define RN     2000
#define RM_F   100.0f
#define LCH    4        // LMAX+1
#define BD     4
#define SPLIT  16       // eigenvalue chunks per (b,l) matrix
#define EIG_PER_BLK (RN / SPLIT)   // 125
#define NTHR   128      // 4 wave32s
#define NITER  40       // bisection iterations (fp32-saturating)

typedef uint32_t u32x4 __attribute__((ext_vector_type(4)));
typedef uint32_t u32x8 __attribute__((ext_vector_type(8)));

// Eigenvalues of symmetric tridiagonal H (diag varies, offdiag = -s) via
// Sturm-sequence bisection. One block handles 125 eigenvalue indices of one
// (b,l) channel. The ptl row is DMA'd into LDS by the Tensor Data Mover.
__global__ __launch_bounds__(NTHR)
void EvalEig_sturm_kernel(const float* __restrict__ ptl, float* __restrict__ out)
{
    __shared__ __align__(16) float lds_ptl[RN];
    __shared__ __align__(16) float lds_diag[RN];

    const int blk   = blockIdx.x;
    const int bl    = blk / SPLIT;      // 0..15 : (b,l) channel
    const int chunk = blk % SPLIT;      // 0..15 : eigenvalue chunk
    const int b     = bl / LCH;
    const int l     = bl % LCH;
    const int tid   = threadIdx.x;

    // ---- TDM: DMA ptl[b, 0:2000] (8000 B) global -> LDS -------------------
    // Issued once by wave 0 (TDM ignores EXEC; tracked by TENSORcnt).
    if (tid < 32) {
        const uint64_t gaddr = (uint64_t)(uintptr_t)(ptl + (size_t)b * RN);
        // Flat LDS address truncated to 32 bits == LDS byte offset (ISA 10.2).
        const uint32_t laddr = (uint32_t)(uintptr_t)(&lds_ptl[0]);

        // D# group 0 (4 SGPRs): count=1 | lds_addr | global_addr[56:0] | type=2
        u32x4 g0;
        g0.x = 1u;                                        // count=1, flags=0
        g0.y = laddr;                                     // lds_addr
        g0.z = (uint32_t)(gaddr & 0xffffffffu);           // global_addr[31:0]
        g0.w = (uint32_t)((gaddr >> 32) & 0x01ffffffu)    // global_addr[56:32]
             | (2u << 30);                                // type = 2 ("image")

        // D# group 1 (8 SGPRs): 2D tensor 2000x1, tile 2000x1, 4-byte elems
        u32x8 g1;
        g1.s0 = (2u << 16);                   // data_size = 2 (4 bytes)
        g1.s1 = ((uint32_t)RN) << 16;         // tensor_dim0[15:0] = 2000
        g1.s2 = (1u << 16);                   // tensor_dim0 hi=0; tensor_dim1=1
        g1.s3 = ((uint32_t)RN) << 16;         // tensor_dim1 hi=0; tile_dim0=2000
        g1.s4 = 1u;                           // tile_dim1=1; tile_dim2=0
        g1.s5 = (uint32_t)RN;                 // tensor_dim0_stride[31:0] = 2000
        g1.s6 = ((uint32_t)RN) << 16;         // stride0 hi=0; tensor_dim1_stride lo
        g1.s7 = 0u;                           // tensor_dim1_stride hi

        asm volatile("tensor_load_to_lds %0, %1" :: "s"(g0), "s"(g1) : "memory");
#if __has_builtin(__builtin_amdgcn_s_wait_tensorcnt)
        __builtin_amdgcn_s_wait_tensorcnt(0);
#else
        asm volatile("s_wait_tensorcnt 0x0" ::: "memory");
#endif
    }
    __syncthreads();

    // ---- Build diagonal in LDS -------------------------------------------
    // s = (RN/RM)^2 = 400 ; diag_i = 2*s + ptl_i + l(l+1)/r_i^2
    const float s   = (float)RN / RM_F * ((float)RN / RM_F);    // 400
    const float e2  = s * s;                                    // offdiag^2
    const float ll1 = (float)(l * (l + 1));
    const float r0  = RM_F / (float)RN;                         // 0.05
    const float dr  = (RM_F - r0) / (float)(RN - 1);

    for (int i = tid; i < RN; i += NTHR) {
        const float r = r0 + dr * (float)i;
        lds_diag[i] = 2.0f * s + lds_ptl[i] + ll1 / (r * r);
    }
    __syncthreads();

    const float4* __restrict__ diag4 = (const float4*)lds_diag;

    // ---- Gershgorin bounds (uniform broadcast scan, ds_load_b128) ---------
    float dmin =  3.0e38f, dmax = -3.0e38f;
    for (int i4 = 0; i4 < RN / 4; ++i4) {
        const float4 v = diag4[i4];
        dmin = fminf(dmin, fminf(fminf(v.x, v.y), fminf(v.z, v.w)));
        dmax = fmaxf(dmax, fmaxf(fmaxf(v.x, v.y), fmaxf(v.z, v.w)));
    }
    float lo = dmin - 2.0f * s - 1.0f;
    float hi = dmax + 2.0f * s + 1.0f;

    // ---- Bisection: thread -> eigenvalue index k -------------------------
    const bool valid = (tid < EIG_PER_BLK);
    const int  k     = chunk * EIG_PER_BLK + (valid ? tid : 0);

    for (int it = 0; it < NITER; ++it) {
        const float x = 0.5f * (lo + hi);
        // Sturm count: # negative LDL^T pivots == # eigenvalues < x.
        // Seed d huge so e2*rcp(d) ~ 0 for the first pivot (no special case).
        // Zero pivots: rcp(0)=inf -> d=-inf (counts negative, LAPACK-style),
        // then rcp(-inf)=-0 recovers the chain; no NaN path exists.
        float d = 3.0e37f;
        int cnt = 0;
        for (int i4 = 0; i4 < RN / 4; ++i4) {
            const float4 v = diag4[i4];                       // ds_load_b128
            d = (v.x - x) - e2 * __builtin_amdgcn_rcpf(d);  cnt += (d < 0.0f);
            d = (v.y - x) - e2 * __builtin_amdgcn_rcpf(d);  cnt += (d < 0.0f);
            d = (v.z - x) - e2 * __builtin_amdgcn_rcpf(d);  cnt += (d < 0.0f);
            d = (v.w - x) - e2 * __builtin_amdgcn_rcpf(d);  cnt += (d < 0.0f);
        }
        if (cnt > k) hi = x; else lo = x;
    }

    if (valid)
        out[((size_t)(b * LCH + l)) * RN + k] = 0.5f * (lo + hi);
}

extern "C" void kernel_launch(void* const* d_in, const int* in_sizes, int n_in,
                              void* d_out, int out_size, void* d_ws, size_t ws_size,
                              hipStream_t stream) {
    (void)in_sizes; (void)n_in; (void)d_ws; (void)ws_size; (void)out_size;
    const float* ptl = (const float*)d_in[0];
    float* out = (float*)d_out;
    // 16 channels x 16 chunks = 256 blocks; 128 threads = 4 wave32s each.
    EvalEig_sturm_kernel<<<dim3(BD * LCH * SPLIT), dim3(NTHR), 0, stream>>>(ptl, out);
}